// GlobalTUL_46986942218300
// MI455X (gfx1250) — compile-verified
//
#include <hip/hip_runtime.h>
#include <hip/hip_bf16.h>

// ---------------------------------------------------------------------------
// fp32 WMMA types (CDNA5 / gfx1250, wave32)
// ---------------------------------------------------------------------------
typedef float v2f __attribute__((ext_vector_type(2)));
typedef float v8f __attribute__((ext_vector_type(8)));

#define KH        128   // hidden size (K dim of every GEMM here)
#define XS_STRIDE 132   // 128 + 4 pad -> LDS bank-conflict free A reads
#define RT        4     // row tiles (of 16) per block -> 64 rows staged in LDS

// ---------------------------------------------------------------------------
// elementwise helpers
// ---------------------------------------------------------------------------
__global__ void zero_f32_kernel(float* __restrict__ p, int n) {
    int i = blockIdx.x * blockDim.x + threadIdx.x;
    if (i < n) p[i] = 0.0f;
}

__global__ void deg_kernel(const int* __restrict__ col,
                           const float* __restrict__ w,
                           float* __restrict__ deg, int E) {
    int e = blockIdx.x * blockDim.x + threadIdx.x;
    if (e < E) {
        __hip_atomic_fetch_add(&deg[col[e]], w[e],
                               __ATOMIC_RELAXED, __HIP_MEMORY_SCOPE_AGENT);
    }
}

__global__ void rsqrt_kernel(float* __restrict__ deg, int n) {
    int i = blockIdx.x * blockDim.x + threadIdx.x;
    if (i < n) {
        float d = deg[i];
        deg[i] = (d > 0.0f) ? rsqrtf(fmaxf(d, 1e-30f)) : 0.0f;   // deg -> dis
    }
}

__global__ void norm_kernel(const int* __restrict__ ei,   // [2*E], row then col
                            const float* __restrict__ w,
                            const float* __restrict__ dis,
                            float* __restrict__ nrm, int E) {
    int e = blockIdx.x * blockDim.x + threadIdx.x;
    if (e < E) {
        int r = ei[e];
        int c = ei[E + e];
        nrm[e] = dis[r] * w[e] * dis[c];
    }
}

// target buffer starts as broadcast bias, scatter adds on top
__global__ void fill_bias_kernel(float* __restrict__ buf,
                                 const float* __restrict__ bias,
                                 int total, int H) {
    int i = blockIdx.x * blockDim.x + threadIdx.x;
    if (i < total) buf[i] = bias[i % H];
}

// out[col[e]] += h[row[e]] * norm[e]
// 256-thread block = 8 edges; each lane handles a float4 of features.
__global__ __launch_bounds__(256)
void scatter_kernel(const int* __restrict__ ei,
                    const float* __restrict__ nrm,
                    const float* __restrict__ h,
                    float* __restrict__ out, int E, int H) {
    int e = blockIdx.x * 8 + (threadIdx.x >> 5);
    int q = (threadIdx.x & 31) * 4;              // feature quad (H=128 -> full)
    if (e < E && q < H) {
        int   r = ei[e];
        int   c = ei[E + e];
        float s = nrm[e];
        const float4 v = *(const float4*)(h + (size_t)r * H + q);
        float* o = out + (size_t)c * H + q;
        __hip_atomic_fetch_add(o + 0, v.x * s, __ATOMIC_RELAXED, __HIP_MEMORY_SCOPE_AGENT);
        __hip_atomic_fetch_add(o + 1, v.y * s, __ATOMIC_RELAXED, __HIP_MEMORY_SCOPE_AGENT);
        __hip_atomic_fetch_add(o + 2, v.z * s, __ATOMIC_RELAXED, __HIP_MEMORY_SCOPE_AGENT);
        __hip_atomic_fetch_add(o + 3, v.w * s, __ATOMIC_RELAXED, __HIP_MEMORY_SCOPE_AGENT);
    }
}

// ---------------------------------------------------------------------------
// fp32 WMMA GEMM:  out[N x Ncol] = X[N x 128] @ W[128 x Ncol] (+ bias)
//
// Block = 256 threads = 8 waves. Block stages RT*16 = 64 rows of X in LDS;
// wave w owns col-tile (blockIdx.y*8 + w) and accumulates RT 16x16 tiles,
// reusing each B fragment RT times. Inner loop per K-chunk:
//   4x ds_load_b64 (A) + 2x global_load_b32 (B) + 4x V_WMMA_F32_16X16X4_F32
// Branch-free: OOB columns are clamped (garbage stays in OOB lanes, stores
// are predicated), so EXEC stays all-ones around the WMMAs.
//
// 16x16x4 f32 fragment layout (ISA 7.12.2):
//   A (16x4):  lane l -> M = l%16 ; vgpr v -> K = k0 + (l<16?0:2) + v
//   B (4x16):  lane l -> N = l%16 ; vgpr v -> K = k0 + (l<16?0:2) + v
//   C/D:       lane l -> N = l%16 ; vgpr v -> M = v + (l<16?0:8)
// ---------------------------------------------------------------------------
__global__ __launch_bounds__(256)
void gemm_f32_wmma_kernel(const float* __restrict__ X,
                          const float* __restrict__ W,
                          const float* __restrict__ bias,   // may be null
                          float* __restrict__ out,
                          int Nrows, int Ncol) {
    __shared__ float xs[RT * 16 * XS_STRIDE];                // ~33.8 KB

    const int lane    = threadIdx.x & 31;
    const int wave    = threadIdx.x >> 5;
    const int rowBase = blockIdx.x * (RT * 16);

    // ---- stage X tile (64 x 128 fp32) into LDS as float4 ------------------
    const int nvec = RT * 16 * (KH / 4);                     // 2048 slots
    for (int s = threadIdx.x; s < nvec; s += 256) {
        int r  = s >> 5;                                     // row in tile
        int c4 = s & 31;                                     // float4 column
        int gr = rowBase + r;
        float4 v = make_float4(0.f, 0.f, 0.f, 0.f);
        if (gr < Nrows)
            v = *(const float4*)(X + (size_t)gr * KH + (size_t)c4 * 4);
        *(float4*)(&xs[r * XS_STRIDE + c4 * 4]) = v;
    }
    __syncthreads();

    // ---- per-wave tile coordinates ----------------------------------------
    const int  colTile = blockIdx.y * 8 + wave;
    const int  n       = colTile * 16 + (lane & 15);
    const bool nOK     = (n < Ncol);
    const int  nc      = nOK ? n : (Ncol - 1);               // clamp, no branch
    const int  half    = lane >> 4;                          // 0 | 1
    const int  koff    = half * 2;
    const int  mA      = lane & 15;

    // B pointers: two rows (k, k+1) of the lane-half's K offset, stride 4*Ncol
    const float* wp0 = W + (size_t)koff * Ncol + nc;
    const float* wp1 = wp0 + Ncol;
    const size_t wstep = (size_t)4 * Ncol;

    const float* ap = &xs[mA * XS_STRIDE + koff];            // LDS base

    v8f acc0 = {}, acc1 = {}, acc2 = {}, acc3 = {};

    // ---- K loop: 4 WMMAs per chunk, B fragment reused 4x ------------------
    for (int kk = 0; kk < KH; kk += 4) {
        v2f b;
        b.x = *wp0;
        b.y = *wp1;
        wp0 += wstep;
        wp1 += wstep;

        v2f a0 = *(const v2f*)(ap + 0 * 16 * XS_STRIDE + kk);
        v2f a1 = *(const v2f*)(ap + 1 * 16 * XS_STRIDE + kk);
        v2f a2 = *(const v2f*)(ap + 2 * 16 * XS_STRIDE + kk);
        v2f a3 = *(const v2f*)(ap + 3 * 16 * XS_STRIDE + kk);

        acc0 = __builtin_amdgcn_wmma_f32_16x16x4_f32(false, a0, false, b, (short)0, acc0, false, false);
        acc1 = __builtin_amdgcn_wmma_f32_16x16x4_f32(false, a1, false, b, (short)0, acc1, false, false);
        acc2 = __builtin_amdgcn_wmma_f32_16x16x4_f32(false, a2, false, b, (short)0, acc2, false, false);
        acc3 = __builtin_amdgcn_wmma_f32_16x16x4_f32(false, a3, false, b, (short)0, acc3, false, false);
    }

    // ---- store (optionally fused bias) ------------------------------------
    float bv = (bias != nullptr) ? bias[nc] : 0.0f;
#pragma unroll
    for (int t = 0; t < RT; ++t) {
        const v8f& acc = (t == 0) ? acc0 : (t == 1) ? acc1 : (t == 2) ? acc2 : acc3;
#pragma unroll
        for (int v = 0; v < 8; ++v) {
            int m = rowBase + t * 16 + v + half * 8;
            if (nOK && m < Nrows)
                out[(size_t)m * Ncol + n] = acc[v] + bv;
        }
    }
}

// ---------------------------------------------------------------------------
// launch
// ---------------------------------------------------------------------------
extern "C" void kernel_launch(void* const* d_in, const int* in_sizes, int n_in,
                              void* d_out, int out_size, void* d_ws, size_t ws_size,
                              hipStream_t stream) {
    const int E  = in_sizes[1];          // 640000
    const int H  = in_sizes[4];          // 128
    const int N  = in_sizes[2] / H;      // 100000
    const int NU = in_sizes[8];          // 1000

    const int*   ei  = (const int*)d_in[0];     // [2,E]: rows then cols
    const float* w   = (const float*)d_in[1];
    const float* emb = (const float*)d_in[2];
    const float* W1  = (const float*)d_in[3];
    const float* b1  = (const float*)d_in[4];
    const float* W2  = (const float*)d_in[5];
    const float* b2  = (const float*)d_in[6];
    const float* Wp  = (const float*)d_in[7];
    const float* bp  = (const float*)d_in[8];
    float* out = (float*)d_out;

    // workspace layout (~106 MB)
    float* deg  = (float*)d_ws;                  // N   (becomes dis in place)
    float* nrm  = deg  + N;                      // E
    float* bufA = nrm  + E;                      // N*H
    float* bufB = bufA + (size_t)N * H;          // N*H

    const int T  = 256;
    const int NH = N * H;

    // --- symmetric normalization coefficients ------------------------------
    zero_f32_kernel<<<(N + T - 1) / T, T, 0, stream>>>(deg, N);
    deg_kernel<<<(E + T - 1) / T, T, 0, stream>>>(ei + E, w, deg, E);
    rsqrt_kernel<<<(N + T - 1) / T, T, 0, stream>>>(deg, N);
    norm_kernel<<<(E + T - 1) / T, T, 0, stream>>>(ei, w, deg, nrm, E);

    const int rowBlocks = (N + RT * 16 - 1) / (RT * 16);
    const dim3 gH(rowBlocks, 1);                             // Ncol=128 -> 8 tiles
    const dim3 gP(rowBlocks, (((NU + 15) / 16) + 7) / 8);    // Ncol=1000 -> 63 tiles
    const int scatterBlocks = (E + 7) / 8;

    // --- layer 1: h1 = scatter(emb @ W1) + b1 ------------------------------
    gemm_f32_wmma_kernel<<<gH, T, 0, stream>>>(emb, W1, nullptr, bufA, N, H);
    fill_bias_kernel<<<(NH + T - 1) / T, T, 0, stream>>>(bufB, b1, NH, H);
    scatter_kernel<<<scatterBlocks, T, 0, stream>>>(ei, nrm, bufA, bufB, E, H);

    // --- layer 2: h2 = scatter(h1 @ W2) + b2 -------------------------------
    gemm_f32_wmma_kernel<<<gH, T, 0, stream>>>(bufB, W2, nullptr, bufA, N, H);
    fill_bias_kernel<<<(NH + T - 1) / T, T, 0, stream>>>(bufB, b2, NH, H);
    scatter_kernel<<<scatterBlocks, T, 0, stream>>>(ei, nrm, bufA, bufB, E, H);

    // --- predictions = h2 @ Wp + bp ----------------------------------------
    gemm_f32_wmma_kernel<<<gP, T, 0, stream>>>(bufB, Wp, bp, out, N, NU);
}